// PowerLayer_88751204205099
// MI455X (gfx1250) — compile-verified
//
#include <hip/hip_runtime.h>
#include <stdint.h>

#define EPSF 1e-9f

// Tile configuration (reference shapes: B=1024, D=512, U=512 tile exactly).
constexpr int BT  = 64;       // x rows per block
constexpr int UT  = 32;       // w/p cols per block
constexpr int DK  = 32;       // reduction chunk
constexpr int SLS = BT + 4;   // padded stride for transposed L tile: keeps 16B
                              // alignment, kills bank conflicts on transposed writes
constexpr int WWO = UT * 2;   // floats per d-row of interleaved {w0,w1,wo0,wo1}

// ---- CDNA5 async global->LDS copies (tracked by ASYNCcnt) ----
__device__ __forceinline__ void async_load_b128_to_lds(uint32_t lds_byte_off,
                                                       const void* gaddr) {
  asm volatile("global_load_async_to_lds_b128 %0, %1, off"
               :: "v"(lds_byte_off), "v"((unsigned long long)(uintptr_t)gaddr)
               : "memory");
}
__device__ __forceinline__ void async_load_b64_to_lds(uint32_t lds_byte_off,
                                                      const void* gaddr) {
  asm volatile("global_load_async_to_lds_b64 %0, %1, off"
               :: "v"(lds_byte_off), "v"((unsigned long long)(uintptr_t)gaddr)
               : "memory");
}
// Drain this wave's async copies down to <= N outstanding (in-order completion).
__device__ __forceinline__ void wait_async_le3() {
  asm volatile("s_wait_asynccnt 3" ::: "memory");
}
__device__ __forceinline__ void wait_async_le0() {
  asm volatile("s_wait_asynccnt 0" ::: "memory");
}

// LDS byte offset of a __shared__ object: flat LDS addresses are
// {aperture, offset}, so truncation to 32 bits yields the DS offset.
__device__ __forceinline__ uint32_t lds_off(const void* shared_ptr) {
  return (uint32_t)(uintptr_t)shared_ptr;
}

__global__ __launch_bounds__(256)
void powerlayer_fused_kernel(const float* __restrict__ x,    // [B, D]
                             const float* __restrict__ w,    // [D, U]
                             const float* __restrict__ p,    // [D, U]
                             const float* __restrict__ bias, // [U]
                             float* __restrict__ out,        // [B, U]
                             int B, int D, int U) {
  // Ping-pong DMA targets; single-buffered L tile (produced by VALU, fenced).
  __shared__ __align__(16) float    sL  [DK * SLS];     // log2|x+eps|, [d][row]
  __shared__ __align__(16) float    sP  [2][DK * UT];   // p tile, [d][col]
  __shared__ __align__(16) float    sWWO[2][DK * WWO];  // [d][colpair]{w0,w1,wo0,wo1}
  __shared__              uint32_t  sNeg[2][DK * 2];    // sign(x+eps) bit per row

  const int tid = threadIdx.x;
  const int tx  = tid & 15;   // 16 col-groups * 2 cols = UT
  const int ty  = tid >> 4;   // 16 row-groups * 4 rows = BT
  const int b0g = blockIdx.x * BT;
  const int u0g = blockIdx.y * UT;

  float acc[4][2] = {};

  // ---- helper lambda: issue this chunk's w/p DMA (3 async instrs per wave) ----
  auto issue_dma = [&](int dbase, int nb) {
    // p tile: DK*UT/4 = 256 b128 packets, one per thread
    {
      const int d  = tid >> 3;           // 8 packets per d-row
      const int u4 = (tid & 7) << 2;
      async_load_b128_to_lds(lds_off(&sP[nb][d * UT + u4]),
                             &p[(size_t)(dbase + d) * U + u0g + u4]);
    }
    // w tile: DK*UT/2 = 512 b64 packets into the low half of each float4 slot
#pragma unroll
    for (int k = 0; k < 2; ++k) {
      const int idx = tid + k * 256;
      const int d   = idx >> 4;          // 16 packets per d-row
      const int q   = idx & 15;          // column pair
      async_load_b64_to_lds(lds_off(&sWWO[nb][d * WWO + q * 4]),
                            &w[(size_t)(dbase + d) * U + u0g + q * 2]);
    }
  };

  // ---- prologue: zero sign buffer 0, kick chunk 0 DMA ----
  if (tid < DK * 2) sNeg[0][tid] = 0u;
  issue_dma(0, 0);

  for (int d0 = 0; d0 < D; d0 += DK) {
    const int c    = d0 / DK;
    const int buf  = c & 1;
    const int nbuf = buf ^ 1;
    const bool has_next = (d0 + DK) < D;

    __syncthreads();  // chunk c-1 compute done: sL / sWWO[nbuf] / sP[nbuf] /
                      // sNeg[nbuf] are free; prologue zeroing visible for c==0

    if (has_next) issue_dma(d0 + DK, nbuf);
    if (tid < DK * 2) sNeg[nbuf][tid] = 0u;   // pre-zero next chunk's sign bits

    // ---- stage x -> L = log2|x+eps| (transposed) + packed sign bits,
    //      overlapping the in-flight async DMA ----
#pragma unroll
    for (int k = 0; k < 2; ++k) {
      const int idx = tid + k * 256;     // BT*DK/4 = 512 float4 packets
      const int row = idx >> 3;          // DK/4 = 8 packets per row
      const int d4  = (idx & 7) << 2;
      const float4 xv = *(const float4*)&x[(size_t)(b0g + row) * D + d0 + d4];
      const float xe[4] = {xv.x + EPSF, xv.y + EPSF, xv.z + EPSF, xv.w + EPSF};
#pragma unroll
      for (int j = 0; j < 4; ++j) {
        const int dd = d4 + j;
        sL[dd * SLS + row] = __builtin_amdgcn_logf(__builtin_fabsf(xe[j]));
        if (xe[j] < 0.0f)
          atomicOr(&sNeg[buf][dd * 2 + (row >> 5)], 1u << (row & 31));
      }
    }

    // Drain only chunk c's 3 async copies (chunk c+1's stay in flight).
    if (has_next) wait_async_le3(); else wait_async_le0();
    __syncthreads();  // all waves: DMA c landed + L/sign staging visible

    // ---- fill wo-slots: wodd = (jnp.mod(p,2)!=0) ? -w : w
    //      (p/2 integral <=> floor-mod(p,2)==0) ----
#pragma unroll
    for (int k = 0; k < 2; ++k) {
      const int idx  = tid + k * 256;    // DK*UT/2 = 512 column pairs
      const int d    = idx >> 4;
      const int q    = idx & 15;
      const int base = d * WWO + q * 4;
      const float2 w2 = *(const float2*)&sWWO[buf][base];
      const float2 p2 = *(const float2*)&sP[buf][d * UT + q * 2];
      const float h0 = 0.5f * p2.x, h1 = 0.5f * p2.y;
      float2 wo;
      wo.x = (__builtin_floorf(h0) != h0) ? -w2.x : w2.x;
      wo.y = (__builtin_floorf(h1) != h1) ? -w2.y : w2.y;
      *(float2*)&sWWO[buf][base + 2] = wo;
    }

    __syncthreads();  // wo-slots visible

    // ---- main compute: 8 outputs / thread / d iteration ----
#pragma unroll 4
    for (int d = 0; d < DK; ++d) {
      const float4   L4 = *(const float4*)&sL[d * SLS + ty * 4];
      const uint32_t nb = sNeg[buf][d * 2 + (ty >> 3)];
      const float2   p2 = *(const float2*)&sP[buf][d * UT + tx * 2];
      const float4   wq = *(const float4*)&sWWO[buf][d * WWO + tx * 4];
      const float Ls[4] = {L4.x, L4.y, L4.z, L4.w};
#pragma unroll
      for (int i = 0; i < 4; ++i) {
        const bool  neg = (nb >> ((ty * 4 + i) & 31)) & 1u;
        const float wa  = neg ? wq.z : wq.x;                     // v_cndmask
        const float wb  = neg ? wq.w : wq.y;
        const float va  = __builtin_amdgcn_exp2f(p2.x * Ls[i]);  // v_exp_f32
        const float vb  = __builtin_amdgcn_exp2f(p2.y * Ls[i]);
        acc[i][0] = __builtin_fmaf(wa, va, acc[i][0]);
        acc[i][1] = __builtin_fmaf(wb, vb, acc[i][1]);
      }
    }
  }

  // ---- epilogue: add bias, store ----
  const int gu = u0g + tx * 2;
  const float bb0 = bias[gu];
  const float bb1 = bias[gu + 1];
#pragma unroll
  for (int i = 0; i < 4; ++i) {
    const int row = b0g + ty * 4 + i;
    float2 o;
    o.x = acc[i][0] + bb0;
    o.y = acc[i][1] + bb1;
    *(float2*)&out[(size_t)row * U + gu] = o;
  }
}

extern "C" void kernel_launch(void* const* d_in, const int* in_sizes, int n_in,
                              void* d_out, int out_size, void* d_ws, size_t ws_size,
                              hipStream_t stream) {
  const float* x    = (const float*)d_in[0];
  const float* w    = (const float*)d_in[1];
  const float* p    = (const float*)d_in[2];
  const float* bias = (const float*)d_in[3];
  float* out        = (float*)d_out;

  const int U = in_sizes[3];           // 512
  const int D = in_sizes[1] / U;       // 512
  const int B = in_sizes[0] / D;       // 1024

  dim3 grid(B / BT, U / UT);           // (16, 16) for the reference shapes
  powerlayer_fused_kernel<<<grid, 256, 0, stream>>>(x, w, p, bias, out, B, D, U);
}